// AggerateLayer_4449586119504
// MI455X (gfx1250) — compile-verified
//
#include <hip/hip_runtime.h>

typedef __attribute__((ext_vector_type(2))) float v2f;
typedef __attribute__((ext_vector_type(8))) float v8f;

#define NCAPS 8
#define H     64
#define FAN   512            // NCAPS * H
#define NPB   16             // nodes per block (WMMA M dim)
#define THREADS 256          // 8 waves of 32

__global__ __launch_bounds__(THREADS)
void agg_gemm_relu_kernel(const float* __restrict__ x,
                          const float* __restrict__ W,
                          const float* __restrict__ bias,
                          const int*   __restrict__ nbr_node,
                          const int*   __restrict__ nbr_rel,
                          float*       __restrict__ out,
                          int N, int M)
{
    __shared__ float acc[NPB * FAN];   // 32 KB: aggregated rows for 16 nodes

    const int tid       = threadIdx.x;
    const int node_base = blockIdx.x * NPB;

    // ---- Phase 1: acc = x (residual term); zero for tail nodes ----
    for (int f = tid; f < NPB * FAN; f += THREADS) {
        const int nl   = f >> 9;          // f / 512
        const int c    = f & (FAN - 1);
        const int node = node_base + nl;
        acc[f] = (node < N) ? x[(size_t)node * FAN + c] : 0.0f;
    }
    __syncthreads();

    // ---- Phase 2: gather neighbor rows + scatter-add into LDS ----
    // 4 groups of 64 threads; each group processes one (node, m) pair per
    // iteration: 64 lanes cover the H=64 row, LDS float atomics resolve
    // relation collisions between concurrent groups.
    {
        const int grp = tid >> 6;         // 0..3
        const int l64 = tid & 63;         // 0..63 (h index)
        const int npairs = NPB * M;
        for (int p = grp; p < npairs; p += 4) {
            const int nl   = p / M;
            const int m    = p - nl * M;
            const int node = node_base + nl;
            if (node < N) {
                const int e   = node * M + m;
                const int nbr = nbr_node[e];   // in [0, N]; N == zero row
                const int rel = nbr_rel[e];    // in [0, NCAPS)
                if (nbr < N) {
                    const float v = x[((size_t)nbr * NCAPS + rel) * H + l64];
                    atomicAdd(&acc[nl * FAN + rel * H + l64], v);  // ds_add_f32
                }
            }
        }
    }
    __syncthreads();

    // ---- Phase 3: out[16 x 512] = relu(acc[16 x 512] @ W^T + b) ----
    // Wave w owns 4 column tiles: cols [w*64, w*64+64). K in steps of 4 using
    // V_WMMA_F32_16X16X4_F32.
    const int wave = tid >> 5;            // 0..7
    const int lane = tid & 31;
    const int m16  = lane & 15;           // A/B/C minor index
    const int hi   = lane >> 4;           // 0/1: K phase for A/B, M phase for C/D

    // A-fragment source: row m16 of acc. Layout: vgpr0 = A[m16, k0+2*hi],
    // vgpr1 = A[m16, k0+2*hi+1]  -> contiguous float2 in LDS.
    const float* arow = &acc[m16 * FAN];

    // B-fragment sources: B[k, n] = W[n, k] (W row-major 512x512), so lane
    // needs W[col, k0+2*hi .. +1] -> contiguous float2 in global (L2-resident).
    const int col0 = (wave * 4 + 0) * 16 + m16;
    const int col1 = (wave * 4 + 1) * 16 + m16;
    const int col2 = (wave * 4 + 2) * 16 + m16;
    const int col3 = (wave * 4 + 3) * 16 + m16;
    const float* w0 = W + (size_t)col0 * FAN;
    const float* w1 = W + (size_t)col1 * FAN;
    const float* w2 = W + (size_t)col2 * FAN;
    const float* w3 = W + (size_t)col3 * FAN;

    v8f c0 = {}, c1 = {}, c2 = {}, c3 = {};

    #pragma unroll 4
    for (int k0 = 0; k0 < FAN; k0 += 4) {
        const int ka = k0 + 2 * hi;
        const v2f a  = *(const v2f*)(arow + ka);   // ds_load_b64
        const v2f b0 = *(const v2f*)(w0 + ka);     // global_load_b64
        const v2f b1 = *(const v2f*)(w1 + ka);
        const v2f b2 = *(const v2f*)(w2 + ka);
        const v2f b3 = *(const v2f*)(w3 + ka);
        c0 = __builtin_amdgcn_wmma_f32_16x16x4_f32(false, a, false, b0,
                                                   (short)0, c0, false, false);
        c1 = __builtin_amdgcn_wmma_f32_16x16x4_f32(false, a, false, b1,
                                                   (short)0, c1, false, false);
        c2 = __builtin_amdgcn_wmma_f32_16x16x4_f32(false, a, false, b2,
                                                   (short)0, c2, false, false);
        c3 = __builtin_amdgcn_wmma_f32_16x16x4_f32(false, a, false, b3,
                                                   (short)0, c3, false, false);
    }

    // ---- Epilogue: bias + ReLU + store. C/D layout: element v of the v8f is
    // D[M = v + 8*hi, N = m16]; divergence here is fine (WMMA is done).
    #define STORE_TILE(CC, COL)                                          \
    {                                                                    \
        const float bv = bias[(COL)];                                    \
        _Pragma("unroll")                                                \
        for (int v = 0; v < 8; ++v) {                                    \
            const int node = node_base + v + 8 * hi;                     \
            if (node < N) {                                              \
                float val = (CC)[v] + bv;                                \
                out[(size_t)node * FAN + (COL)] = val > 0.0f ? val : 0.0f; \
            }                                                            \
        }                                                                \
    }

    STORE_TILE(c0, col0)
    STORE_TILE(c1, col1)
    STORE_TILE(c2, col2)
    STORE_TILE(c3, col3)
    #undef STORE_TILE
}

extern "C" void kernel_launch(void* const* d_in, const int* in_sizes, int n_in,
                              void* d_out, int out_size, void* d_ws, size_t ws_size,
                              hipStream_t stream)
{
    const float* x    = (const float*)d_in[0];
    const float* W    = (const float*)d_in[1];
    const float* bias = (const float*)d_in[2];
    const int*   nn   = (const int*)d_in[3];
    const int*   nr   = (const int*)d_in[4];
    float*       out  = (float*)d_out;

    const int N = in_sizes[0] / (NCAPS * H);
    const int M = in_sizes[3] / N;

    const int nblocks = (N + NPB - 1) / NPB;
    agg_gemm_relu_kernel<<<nblocks, THREADS, 0, stream>>>(x, W, bias, nn, nr,
                                                          out, N, M);
}